// BaselineModel_75746043232309
// MI455X (gfx1250) — compile-verified
//
#include <hip/hip_runtime.h>

typedef float v2f __attribute__((ext_vector_type(2)));
typedef float v4f __attribute__((ext_vector_type(4)));
typedef float v8f __attribute__((ext_vector_type(8)));

#define B_ 64
#define T_ 512
#define I_ 64
#define H_ 1024
#define O_ 64

// D = A(16x4 f32) * B(4x16 f32) + C(16x16 f32), full fp32 precision.
// 8-arg form: (neg_a, A, neg_b, B, c_mod, C, reuse_a, reuse_b)
__device__ __forceinline__ v8f wmma_f32(v2f a, v2f b, v8f c) {
    return __builtin_amdgcn_wmma_f32_16x16x4_f32(false, a, false, b, (short)0, c,
                                                 false, false);
}

// Process 8 physical K-columns with one b128 load per operand and two WMMAs.
// K-permutation: lane-half kh supplies phys cols {k8+4kh .. k8+4kh+3}; the "lo"
// WMMA reduces cols {k8,k8+1}∪{k8+4,k8+5}, the "hi" one {k8+2,k8+3}∪{k8+6,k8+7}.
// A and B fragments use identical indexing, so the permutation is consistent.
__device__ __forceinline__ void mma_k8(const v4f a4, const v4f b4, v8f& acc) {
    v2f alo = {a4.x, a4.y}, ahi = {a4.z, a4.w};
    v2f blo = {b4.x, b4.y}, bhi = {b4.z, b4.w};
    acc = wmma_f32(alo, blo, acc);
    acc = wmma_f32(ahi, bhi, acc);
}

// ---------------------------------------------------------------------------
// Kernel 1: xw[B*T, H] = x[B*T, I] @ W_ih^T + (b_ih + b_hh)
// One wave per 16(M) x 32(N) strip. grid = (H/32, B*T/16), block = 32.
// ---------------------------------------------------------------------------
__global__ void xw_kernel(const float* __restrict__ x,
                          const float* __restrict__ W_ih,
                          const float* __restrict__ b_ih,
                          const float* __restrict__ b_hh,
                          float* __restrict__ xw) {
    const int lane = threadIdx.x & 31;
    const int l16  = lane & 15;
    const int kh   = lane >> 4;
    const int n0   = blockIdx.x * 32;
    const int m0   = blockIdx.y * 16;
    const int n_a  = n0 + l16;
    const int n_b  = n0 + 16 + l16;

    const float bias_a = b_ih[n_a] + b_hh[n_a];
    const float bias_b = b_ih[n_b] + b_hh[n_b];
    v8f acc0, acc1;
#pragma unroll
    for (int r = 0; r < 8; ++r) { acc0[r] = bias_a; acc1[r] = bias_b; }

    const int m = m0 + l16;
    for (int k8 = 0; k8 < I_; k8 += 8) {
        const int pa = k8 + 4 * kh;
        v4f a4  = *(const v4f*)(x    + (long)m   * I_ + pa);
        v4f b40 = *(const v4f*)(W_ih + (long)n_a * I_ + pa);
        v4f b41 = *(const v4f*)(W_ih + (long)n_b * I_ + pa);
        mma_k8(a4, b40, acc0);
        mma_k8(a4, b41, acc1);
    }

#pragma unroll
    for (int r = 0; r < 8; ++r) {
        const long row = (long)(m0 + r + 8 * kh);
        xw[row * H_ + n_a] = acc0[r];
        xw[row * H_ + n_b] = acc1[r];
    }
}

// ---------------------------------------------------------------------------
// Kernel 2 (run T times, graph-sequenced): in-place
//   hio[b, n] = relu(hio[b, n] + sum_k hprev[b, k] * W_hh[n, k])
// hio points at hidden_list[:, t, :]   (row stride T*H)
// hprev points at hidden_list[:, t-1,:] (row stride T*H) or initial h (stride H)
// grid = (H/32, B/16), block = 32.
// ---------------------------------------------------------------------------
__global__ void rnn_step_kernel(const float* __restrict__ hprev, long lda,
                                float* __restrict__ hio,
                                const float* __restrict__ W_hh) {
    const int lane = threadIdx.x & 31;
    const int l16  = lane & 15;
    const int kh   = lane >> 4;
    const int n0   = blockIdx.x * 32;
    const int m0   = blockIdx.y * 16;
    const int n_a  = n0 + l16;
    const int n_b  = n0 + 16 + l16;
    const long ldc = (long)T_ * H_;

    v8f acc0, acc1;
#pragma unroll
    for (int r = 0; r < 8; ++r) {
        const long row = (long)(m0 + r + 8 * kh);
        acc0[r] = hio[row * ldc + n_a];   // xw_t tile (C input)
        acc1[r] = hio[row * ldc + n_b];
    }

    const int m = m0 + l16;
    for (int k8 = 0; k8 < H_; k8 += 8) {
        const int pa = k8 + 4 * kh;
        v4f a4  = *(const v4f*)(hprev + (long)m   * lda + pa);
        v4f b40 = *(const v4f*)(W_hh  + (long)n_a * H_  + pa);
        v4f b41 = *(const v4f*)(W_hh  + (long)n_b * H_  + pa);
        mma_k8(a4, b40, acc0);
        mma_k8(a4, b41, acc1);
    }

#pragma unroll
    for (int r = 0; r < 8; ++r) {
        const long row = (long)(m0 + r + 8 * kh);
        const float v0 = acc0[r] > 0.f ? acc0[r] : 0.f;
        const float v1 = acc1[r] > 0.f ? acc1[r] : 0.f;
        hio[row * ldc + n_a] = v0;
        hio[row * ldc + n_b] = v1;
    }
}

// ---------------------------------------------------------------------------
// Kernel 3: out[B*T, O] = hs[B*T, H] @ W_out^T + b_out
// grid = (O/32, B*T/16), block = 32.
// ---------------------------------------------------------------------------
__global__ void out_kernel(const float* __restrict__ hs,
                           const float* __restrict__ W_out,
                           const float* __restrict__ b_out,
                           float* __restrict__ out) {
    const int lane = threadIdx.x & 31;
    const int l16  = lane & 15;
    const int kh   = lane >> 4;
    const int n0   = blockIdx.x * 32;
    const int m0   = blockIdx.y * 16;
    const int n_a  = n0 + l16;
    const int n_b  = n0 + 16 + l16;

    const float bias_a = b_out[n_a];
    const float bias_b = b_out[n_b];
    v8f acc0, acc1;
#pragma unroll
    for (int r = 0; r < 8; ++r) { acc0[r] = bias_a; acc1[r] = bias_b; }

    const int m = m0 + l16;
    for (int k8 = 0; k8 < H_; k8 += 8) {
        const int pa = k8 + 4 * kh;
        v4f a4  = *(const v4f*)(hs    + (long)m   * H_ + pa);
        v4f b40 = *(const v4f*)(W_out + (long)n_a * H_ + pa);
        v4f b41 = *(const v4f*)(W_out + (long)n_b * H_ + pa);
        mma_k8(a4, b40, acc0);
        mma_k8(a4, b41, acc1);
    }

#pragma unroll
    for (int r = 0; r < 8; ++r) {
        const long row = (long)(m0 + r + 8 * kh);
        out[row * O_ + n_a] = acc0[r];
        out[row * O_ + n_b] = acc1[r];
    }
}

// h_last[b, h] = hidden_list[b, T-1, h]
__global__ void hlast_kernel(const float* __restrict__ hs,
                             float* __restrict__ hlast) {
    const int i = blockIdx.x * blockDim.x + threadIdx.x;  // 0 .. B*H-1
    const int b = i / H_;
    const int h = i % H_;
    hlast[i] = hs[(long)b * T_ * H_ + (long)(T_ - 1) * H_ + h];
}

extern "C" void kernel_launch(void* const* d_in, const int* in_sizes, int n_in,
                              void* d_out, int out_size, void* d_ws, size_t ws_size,
                              hipStream_t stream) {
    (void)in_sizes; (void)n_in; (void)d_ws; (void)ws_size; (void)out_size;

    const float* x      = (const float*)d_in[0];   // [B,T,I]
    const float* h0     = (const float*)d_in[1];   // [1,B,H] (zeros)
    const float* W_ih   = (const float*)d_in[2];   // [H,I]
    const float* W_hh   = (const float*)d_in[3];   // [H,H]
    const float* b_ih   = (const float*)d_in[4];   // [H]
    const float* b_hh   = (const float*)d_in[5];   // [H]
    const float* W_out  = (const float*)d_in[6];   // [O,H]
    const float* b_out  = (const float*)d_in[7];   // [O]

    float* hs    = (float*)d_out;                       // hidden_list [B,T,H]
    float* outs  = hs + (long)B_ * T_ * H_;             // output_list [B,T,O]
    float* hlast = outs + (long)B_ * T_ * O_;           // h_last [1,B,H]

    // 1) Input projection for all timesteps (stored into the hidden_list slab).
    xw_kernel<<<dim3(H_ / 32, (B_ * T_) / 16), 32, 0, stream>>>(x, W_ih, b_ih, b_hh, hs);

    // 2) Sequential recurrence; each launch is a full [64,1024]x[1024,1024] GEMM
    //    + ReLU, updating hidden_list[:, t, :] in place.
    for (int t = 0; t < T_; ++t) {
        const float* hprev = (t == 0) ? h0 : (hs + (long)(t - 1) * H_);
        const long   lda   = (t == 0) ? (long)H_ : (long)T_ * H_;
        rnn_step_kernel<<<dim3(H_ / 32, B_ / 16), 32, 0, stream>>>(
            hprev, lda, hs + (long)t * H_, W_hh);
    }

    // 3) Output projection over all timesteps.
    out_kernel<<<dim3(O_ / 32, (B_ * T_) / 16), 32, 0, stream>>>(hs, W_out, b_out, outs);

    // 4) h_last copy.
    hlast_kernel<<<(B_ * H_) / 256, 256, 0, stream>>>(hs, hlast);
}